// SparseGather_70222715290213
// MI455X (gfx1250) — compile-verified
//
#include <hip/hip_runtime.h>
#include <stdint.h>

// Sparse block gather for MI455X (gfx1250) via the Tensor Data Mover.
// Input:  (N=4, H=512, W=512, C=64) f32, NHWC.
// For each active block (n, by, bx): copy a 16x16x64 tile (64 KB).
// Per (n,y): the (16 x, 64 c) slab is 4KB contiguous; y-stride = W*C*4 = 128KB.
// => one 2D TDM load (tile 1024 x 16 f32, stride 32768 elems) into LDS,
//    one 2D TDM store (contiguous, non-temporal) from LDS to packed output.
//
// Cache policy rationale: input (256MB) has block-level reuse (4096 random
// draws from 4096 block slots => many repeats) and mostly fits the 192MB L2,
// so loads use default RT. Output is write-once/never-reread => NT store so
// it does not evict reusable input lines from L2.

namespace {
constexpr int BSY = 16, BSX = 16, CH = 64;
constexpr int H = 512, W = 512;
constexpr int ROW_ELEMS     = BSX * CH;            // 1024 f32 contiguous per y
constexpr int BLOCK_ELEMS   = BSY * ROW_ELEMS;     // 16384 f32 = 64 KB
constexpr long long ROW_STRIDE_EL = (long long)W * CH;  // 32768 elems between rows
}

typedef __attribute__((ext_vector_type(4))) uint32_t u32x4;
typedef __attribute__((ext_vector_type(8))) uint32_t u32x8;

// D# group 0 (128b): [1:0]=count=1, [63:32]=lds_addr, [120:64]=global_addr,
//                    [127:126]=type=2 ("image").
__device__ __forceinline__ u32x4 tdm_group0(uint32_t lds_addr, uint64_t gaddr) {
    u32x4 g;
    g.x = 1u;                                      // count=1, user descriptor
    g.y = lds_addr;                                // lds_addr (bytes)
    g.z = (uint32_t)gaddr;                         // global_addr[31:0]
    g.w = ((uint32_t)(gaddr >> 32) & 0x01FFFFFFu)  // global_addr[56:32]
          | (2u << 30);                            // type = 2
    return g;
}

// D# group 1 (256b): [17:16]=data_size(2 -> 4B), [79:48]=tensor_dim0,
// [111:80]=tensor_dim1, [127:112]=tile_dim0, [143:128]=tile_dim1,
// [159:144]=tile_dim2(0), [207:160]=tensor_dim0_stride, rest 0.
__device__ __forceinline__ u32x8 tdm_group1(uint32_t tdim0, uint32_t tdim1,
                                            uint32_t tile0, uint32_t tile1,
                                            uint64_t stride0_el) {
    u32x8 g;
    g.s0 = (2u << 16);                                    // data_size = 4 bytes
    g.s1 = (tdim0 & 0xFFFFu) << 16;                       // tensor_dim0[15:0]
    g.s2 = (tdim0 >> 16) | ((tdim1 & 0xFFFFu) << 16);     // dim0[31:16] | dim1[15:0]
    g.s3 = (tdim1 >> 16) | ((tile0 & 0xFFFFu) << 16);     // dim1[31:16] | tile_dim0
    g.s4 = (tile1 & 0xFFFFu);                             // tile_dim1 | tile_dim2=0
    g.s5 = (uint32_t)stride0_el;                          // tensor_dim0_stride[31:0]
    g.s6 = (uint32_t)(stride0_el >> 32) & 0xFFFFu;        // stride[47:32] | dim1_stride=0
    g.s7 = 0u;
    return g;
}

__global__ void __launch_bounds__(32)
sparse_gather_tdm(const float* __restrict__ in,
                  const int* __restrict__ abi,   // (M,3) int32: n, by, bx
                  float* __restrict__ out,
                  int M)
{
    __shared__ float smem[BLOCK_ELEMS];            // 64 KB staging tile

    const int bid = blockIdx.x;
    if (bid >= M) return;

    // Block coords are uniform per workgroup; force into SGPRs.
    const int n  = __builtin_amdgcn_readfirstlane(abi[3 * bid + 0]);
    const int by = __builtin_amdgcn_readfirstlane(abi[3 * bid + 1]);
    const int bx = __builtin_amdgcn_readfirstlane(abi[3 * bid + 2]);

    const int y0 = by * BSY;
    const int x0 = bx * BSX;

    // Byte address of tile start: &in[n, y0, x0, 0]
    const uint64_t src = (uint64_t)(uintptr_t)in
        + 4ull * ( ((uint64_t)n * H + (uint64_t)y0) * (uint64_t)ROW_STRIDE_EL
                   + (uint64_t)x0 * CH );
    const uint64_t dst = (uint64_t)(uintptr_t)out
        + (uint64_t)bid * (uint64_t)(BLOCK_ELEMS * 4);

    // Low 32 bits of the flat shared address == LDS byte address.
    const uint32_t lds = (uint32_t)(uintptr_t)(&smem[0]);

    // Load descriptor: 2D tile 1024x16 f32, row stride W*C elems (gather side).
    const u32x4 ld_g0 = tdm_group0(lds, src);
    const u32x8 ld_g1 = tdm_group1(ROW_ELEMS, BSY, ROW_ELEMS, BSY,
                                   (uint64_t)ROW_STRIDE_EL);

    // Store descriptor: same 2D tile, fully contiguous (stride = row length).
    const u32x4 st_g0 = tdm_group0(lds, dst);
    const u32x8 st_g1 = tdm_group1(ROW_ELEMS, BSY, ROW_ELEMS, BSY,
                                   (uint64_t)ROW_ELEMS);

    // TDM: global (strided) -> LDS. VADDR2/VADDR3 omitted => 2D descriptor.
    // Default RT temporal hint: input blocks have reuse and ~fit in 192MB L2.
    asm volatile("tensor_load_to_lds %0, %1"
                 :
                 : "s"(ld_g0), "s"(ld_g1)
                 : "memory");

    // Ensure the LDS tile is complete before the TDM store reads it.
    __builtin_amdgcn_s_wait_tensorcnt(0);

    // TDM: LDS -> global (contiguous packed output), non-temporal store:
    // output is write-once, keep it out of L2 so input lines stay resident.
    asm volatile("tensor_store_from_lds %0, %1 th:TH_STORE_NT"
                 :
                 : "s"(st_g0), "s"(st_g1)
                 : "memory");

    // Drain before wave exit (S_ENDPGM also implies wait-idle; belt & braces).
    __builtin_amdgcn_s_wait_tensorcnt(0);
}

extern "C" void kernel_launch(void* const* d_in, const int* in_sizes, int n_in,
                              void* d_out, int out_size, void* d_ws, size_t ws_size,
                              hipStream_t stream) {
    const float* in  = (const float*)d_in[0];   // (4,512,512,64) f32
    // d_in[1] = bin_counts (unused; M recovered from index array size)
    const int*   abi = (const int*)d_in[2];     // (M,3) int32
    float*       out = (float*)d_out;           // (M,16,16,64) f32

    const int M = in_sizes[2] / 3;              // 4096
    if (M <= 0) return;

    sparse_gather_tdm<<<M, 32, 0, stream>>>(in, abi, out, M);
}